// MultiG_pooling_35141422416137
// MI455X (gfx1250) — compile-verified
//
#include <hip/hip_runtime.h>
#include <hip/hip_bf16.h>

#define DDIM   1024
#define NNODES 90
#define BATCH  512
#define KSEL   45
#define ROWS   (BATCH * NNODES)   // 46080

typedef float v2f __attribute__((ext_vector_type(2)));
typedef float v4f __attribute__((ext_vector_type(4)));
typedef float v8f __attribute__((ext_vector_type(8)));

// ---------------------------------------------------------------------------
// Kernel 1: fused pooling scores via V_WMMA_F32_16X16X4_F32.
// 8 waves / block, 16 rows / wave -> 128 rows / block. 360 blocks total.
//
// K-permutation trick: low-half lanes carry global K {k0..k0+3}, high-half
// lanes carry {k0+4..k0+7}, split across two WMMA steps. Since we build the
// B fragment ourselves from LDS with the SAME permutation, the dot product
// is exact, and each lane issues a single b128 load per tensor per 8 K.
// ---------------------------------------------------------------------------
__global__ __launch_bounds__(256) void score_kernel(
    const float* __restrict__ x_sc, const float* __restrict__ x_fc,
    const float* __restrict__ pool_w, const float* __restrict__ multi_w,
    const float* __restrict__ multi_b, float* __restrict__ scores)
{
    __shared__ float s_pw[DDIM];
    __shared__ float s_red[256];

    const int tid = threadIdx.x;

    // Stage pool_w in LDS and compute ||pool_w||^2 with a deterministic tree.
    float ss = 0.f;
    for (int i = tid; i < DDIM; i += 256) {
        float v = pool_w[i];
        s_pw[i] = v;
        ss += v * v;
    }
    s_red[tid] = ss;
    __syncthreads();
    for (int off = 128; off > 0; off >>= 1) {
        if (tid < off) s_red[tid] += s_red[tid + off];
        __syncthreads();
    }
    const float inv_norm = rsqrtf(s_red[0]);

    const float mw0 = multi_w[0];
    const float mw1 = multi_w[1];

    const int lane  = tid & 31;
    const int wave  = tid >> 5;
    const long rowbase = ((long)blockIdx.x * 8 + wave) * 16;

    const int m  = lane & 15;              // row within 16-row tile
    const int h4 = (lane >> 4) * 4;        // global-K sub-offset per lane half
    const float leadf = (m == 0) ? 1.f : 0.f;  // only lanes 0/16 carry B col 0

    const float* __restrict__ pa_sc = x_sc + (rowbase + m) * (long)DDIM + h4;
    const float* __restrict__ pa_fc = x_fc + (rowbase + m) * (long)DDIM + h4;
    const float* __restrict__ ppw   = s_pw + h4;

    v8f acc_sc = {};
    v8f acc_fc = {};

#pragma unroll 4
    for (int k0 = 0; k0 < DDIM; k0 += 8) {
        // One 16-byte load per tensor per lane covers two WMMA steps.
        v4f a4_sc = *(const v4f*)(pa_sc + k0);
        v4f a4_fc = *(const v4f*)(pa_fc + k0);
        // Matching pool_w quad, masked to zero on non-lead lanes (no branches).
        v4f pw4 = *(const v4f*)(ppw + k0) * leadf;

        v2f aA_sc = {a4_sc[0], a4_sc[1]};
        v2f aB_sc = {a4_sc[2], a4_sc[3]};
        v2f aA_fc = {a4_fc[0], a4_fc[1]};
        v2f aB_fc = {a4_fc[2], a4_fc[3]};
        v2f bA    = {pw4[0],  pw4[1]};
        v2f bB    = {pw4[2],  pw4[3]};

        acc_sc = __builtin_amdgcn_wmma_f32_16x16x4_f32(
            false, aA_sc, false, bA, (short)0, acc_sc, false, false);
        acc_fc = __builtin_amdgcn_wmma_f32_16x16x4_f32(
            false, aA_fc, false, bA, (short)0, acc_fc, false, false);
        acc_sc = __builtin_amdgcn_wmma_f32_16x16x4_f32(
            false, aB_sc, false, bB, (short)0, acc_sc, false, false);
        acc_fc = __builtin_amdgcn_wmma_f32_16x16x4_f32(
            false, aB_fc, false, bB, (short)0, acc_fc, false, false);
    }

    // D column 0: lane 0 holds rows 0-7 in VGPR0-7, lane 16 holds rows 8-15.
    if (m == 0) {
        const int mbase = (lane >> 4) * 8;
#pragma unroll
        for (int i = 0; i < 8; ++i) {
            long row   = rowbase + mbase + i;
            int  node  = (int)(row % NNODES);
            float ssc  = tanhf(acc_sc[i] * inv_norm);
            float sfc  = tanhf(acc_fc[i] * inv_norm);
            float z    = ssc * mw0 + sfc * mw1 + multi_b[node];
            scores[row] = 1.f / (1.f + expf(-z));
        }
    }
}

// ---------------------------------------------------------------------------
// Kernel 2: per-batch stable top-K by rank counting (matches argsort(-s)).
// ---------------------------------------------------------------------------
__global__ __launch_bounds__(128) void topk_kernel(
    const float* __restrict__ scores, int* __restrict__ node_idx)
{
    const int b = blockIdx.x;
    const int t = threadIdx.x;
    __shared__ float s[NNODES];

    if (t < NNODES) s[t] = scores[b * NNODES + t];
    __syncthreads();

    if (t < NNODES) {
        const float si = s[t];
        int rank = 0;
#pragma unroll
        for (int j = 0; j < NNODES; ++j) {
            float sj = s[j];
            if (sj > si || (sj == si && j < t)) ++rank;
        }
        if (rank < KSEL) node_idx[b * KSEL + rank] = t;
    }
}

// ---------------------------------------------------------------------------
// Kernel 3: gather selected rows scaled by score; NT stores (streaming out).
// One block per (b,k); 256 threads x v4f = 1024 floats per tensor.
// ---------------------------------------------------------------------------
__global__ __launch_bounds__(256) void gather_kernel(
    const float* __restrict__ x_sc, const float* __restrict__ x_fc,
    const float* __restrict__ scores, const int* __restrict__ node_idx,
    float* __restrict__ out)
{
    const int bk = blockIdx.x;
    const int b  = bk / KSEL;
    const int n  = node_idx[bk];
    const float w = scores[b * NNODES + n];

    const size_t inoff  = ((size_t)b * NNODES + n) * DDIM;
    const size_t outoff = (size_t)bk * DDIM;
    const size_t fcbase = (size_t)BATCH * KSEL * DDIM;

    const int t4 = threadIdx.x * 4;

    v4f a = *(const v4f*)(x_sc + inoff + t4);
    v4f c = *(const v4f*)(x_fc + inoff + t4);
    a *= w;
    c *= w;

    __builtin_nontemporal_store(a, (v4f*)(out + outoff + t4));
    __builtin_nontemporal_store(c, (v4f*)(out + fcbase + outoff + t4));
}

// ---------------------------------------------------------------------------
extern "C" void kernel_launch(void* const* d_in, const int* in_sizes, int n_in,
                              void* d_out, int out_size, void* d_ws, size_t ws_size,
                              hipStream_t stream)
{
    const float* x_sc    = (const float*)d_in[0];
    const float* x_fc    = (const float*)d_in[1];
    const float* pool_w  = (const float*)d_in[2];
    const float* multi_w = (const float*)d_in[3];
    const float* multi_b = (const float*)d_in[4];
    float* out = (float*)d_out;

    float* scores   = (float*)d_ws;                                       // 46080 f32
    int*   node_idx = (int*)((char*)d_ws + (size_t)ROWS * sizeof(float)); // 512*45 i32

    // 360 blocks x 8 waves x 16 rows = 46080 rows exactly.
    score_kernel<<<ROWS / 128, 256, 0, stream>>>(x_sc, x_fc, pool_w, multi_w,
                                                 multi_b, scores);
    topk_kernel<<<BATCH, 128, 0, stream>>>(scores, node_idx);
    gather_kernel<<<BATCH * KSEL, 256, 0, stream>>>(x_sc, x_fc, scores,
                                                    node_idx, out);
}